// Steamboat_84756884619468
// MI455X (gfx1250) — compile-verified
//
#include <hip/hip_runtime.h>

typedef __attribute__((ext_vector_type(2))) float v2f;
typedef __attribute__((ext_vector_type(8))) float v8f;

#define N_NODES 100000
#define N_NBRS  16
#define D_IN    512
#define DB      32
#define PROJ    128          // 4 * 32 fused projection outputs
#define KCAT    96           // 3 * 32 concatenated score dims
#define SSTR    98           // padded LDS row stride (floats) to avoid bank conflicts
#define INV     (1.0f / 512.0f)
#define INV2    (INV * INV)

__device__ __forceinline__ float elu1(float w) {       // elu(w)+1
    return w > 0.0f ? w + 1.0f : expf(w);
}
__device__ __forceinline__ float sigm(float w) {
    return 1.0f / (1.0f + expf(-w));
}

// ---------------------------------------------------------------- kernel 1
// Global sums of sigmoid(q_local_w), sigmoid(q_global_w); also zero x_bar.
__global__ void Steamboat_sums_kernel(const float* __restrict__ ql,
                                      const float* __restrict__ qg,
                                      float* __restrict__ sm,
                                      float* __restrict__ xb) {
    __shared__ float red0[256], red1[256];
    int t = threadIdx.x;
    float a = 0.0f, b = 0.0f;
    for (int i = t; i < DB * D_IN; i += 256) { a += sigm(ql[i]); b += sigm(qg[i]); }
    red0[t] = a; red1[t] = b;
    __syncthreads();
    for (int s = 128; s > 0; s >>= 1) {
        if (t < s) { red0[t] += red0[t + s]; red1[t] += red1[t + s]; }
        __syncthreads();
    }
    if (t == 0) { sm[0] = red0[0]; sm[1] = red1[0]; }
    if (t < 256) { xb[t] = 0.0f; xb[256 + t] = 0.0f; }
}

// ---------------------------------------------------------------- kernel 2
// Column sums of x into x_bar (divide by N later). grid (2, 128), block 256.
__global__ void Steamboat_mean_kernel(const float* __restrict__ x,
                                      float* __restrict__ xb) {
    int col = blockIdx.x * 256 + threadIdx.x;          // 0..511
    int r0 = blockIdx.y * 782;
    int r1 = r0 + 782; if (r1 > N_NODES) r1 = N_NODES;
    float s = 0.0f;
    for (int r = r0; r < r1; ++r) s += x[(size_t)r * D_IN + col];
    atomicAdd(&xb[col], s);
}

// ---------------------------------------------------------------- kernel 3
// Build Wc[128][512] (transformed, fused projection weights),
// VcatT[512][96] (elu+1 of V matrices, K-contiguous), bias_nn[512].
__global__ void Steamboat_prep_kernel(const float* __restrict__ qk_ego,
                                      const float* __restrict__ ql,
                                      const float* __restrict__ kl,
                                      const float* __restrict__ qg,
                                      const float* __restrict__ ve,
                                      const float* __restrict__ vl,
                                      const float* __restrict__ vg,
                                      const float* __restrict__ bias,
                                      const float* __restrict__ sm,
                                      float* __restrict__ WC,
                                      float* __restrict__ VT,
                                      float* __restrict__ BN) {
    int i = blockIdx.x * blockDim.x + threadIdx.x;
    float inv_sql = 1.0f / sm[0], inv_sqg = 1.0f / sm[1];
    if (i < PROJ * D_IN) {
        int r = i / D_IN, c = i % D_IN;
        float v;
        if      (r < 32) v = qk_ego[r * D_IN + c];
        else if (r < 64) v = sigm(ql[(r - 32) * D_IN + c]) * inv_sql;
        else if (r < 96) v = elu1(kl[(r - 64) * D_IN + c]);
        else             v = sigm(qg[(r - 96) * D_IN + c]) * inv_sqg;
        WC[i] = v;
    } else if (i < PROJ * D_IN + D_IN * KCAT) {
        int j = i - PROJ * D_IN;
        int d = j / KCAT, k = j % KCAT;
        float v;
        if      (k < 32) v = elu1(ve[d * 32 + k]);
        else if (k < 64) v = elu1(vl[d * 32 + (k - 32)]);
        else             v = elu1(vg[d * 32 + (k - 64)]);
        VT[j] = v;
    } else if (i < PROJ * D_IN + D_IN * KCAT + D_IN) {
        int c = i - PROJ * D_IN - D_IN * KCAT;
        BN[c] = elu1(bias[c]);
    }
}

// ---------------------------------------------------------------- kernel 4
// k_glob[c] = (x_bar/N) . elu1(k_global_w[c]) ; 1 wave.
__global__ void Steamboat_kglob_kernel(const float* __restrict__ xb,
                                       const float* __restrict__ kgw,
                                       float* __restrict__ kg) {
    int c = threadIdx.x;                                // 0..31
    const float invN = 1.0f / (float)N_NODES;
    float s = 0.0f;
    for (int d = 0; d < D_IN; ++d) s += xb[d] * invN * elu1(kgw[c * D_IN + d]);
    kg[c] = s;
}

// ---------------------------------------------------------------- kernel 5
// Projection GEMM: P[N][128] = x[N][512] @ Wc^T via V_WMMA_F32_16X16X4_F32.
// 1 wave = 16-node tile x 128 outputs (8 accumulator tiles), K loop of 128.
__global__ __launch_bounds__(256) void Steamboat_proj_kernel(
        const float* __restrict__ x, const float* __restrict__ WC,
        float* __restrict__ P) {
    int wave = threadIdx.x >> 5, lane = threadIdx.x & 31;
    int m = lane & 15, h = lane >> 4;
    int tile = blockIdx.x * 128 + wave * 16;
    int aRow = tile + m; if (aRow > N_NODES - 1) aRow = N_NODES - 1;
    const float* aBase = x + (size_t)aRow * D_IN + 2 * h;
    const float* bBase = WC + (size_t)m * D_IN + 2 * h;

    v8f z = {0, 0, 0, 0, 0, 0, 0, 0};
    v8f acc[8];
#pragma unroll
    for (int g = 0; g < 8; ++g) acc[g] = z;

    for (int j = 0; j < 128; ++j) {
        v2f a = *(const v2f*)(aBase + 4 * j);
#pragma unroll
        for (int g = 0; g < 8; ++g) {
            v2f b = *(const v2f*)(bBase + (size_t)(16 * g) * D_IN + 4 * j);
            acc[g] = __builtin_amdgcn_wmma_f32_16x16x4_f32(
                false, a, false, b, (short)0, acc[g], false, false);
        }
    }
#pragma unroll
    for (int g = 0; g < 8; ++g) {
        int col = 16 * g + m;
#pragma unroll
        for (int r = 0; r < 8; ++r) {
            int row = tile + r + 8 * h;
            if (row < N_NODES) P[(size_t)row * PROJ + col] = acc[g][r];
        }
    }
}

// ---------------------------------------------------------------- kernel 6
// Fused: neighbor gather + scores + normalize into LDS S[16][96],
// then output GEMM out[N][512] = S @ VcatT^T (+bias) via WMMA f32.
__global__ __launch_bounds__(256) void Steamboat_final_kernel(
        const int* __restrict__ adj, const float* __restrict__ P,
        const float* __restrict__ VT, const float* __restrict__ BN,
        const float* __restrict__ KG, float* __restrict__ out) {
    __shared__ __align__(16) float S[8][16 * SSTR];
    int wave = threadIdx.x >> 5, lane = threadIdx.x & 31;
    int m = lane & 15, h = lane >> 4;
    int base = blockIdx.x * 128 + wave * 16;
    float kg = KG[lane];

    // Phase 1: 32 lanes <-> 32 dims, one node at a time.
    for (int i = 0; i < 16; ++i) {
        int node = base + i; if (node > N_NODES - 1) node = N_NODES - 1;
        const float* Pn = P + (size_t)node * PROJ;
        float e  = Pn[lane];
        float es = (e * INV) * (e * INV);
        float q  = Pn[32 + lane];
        int nb = adj[node * N_NBRS + m];            // 16 neighbor ids (dup'd hi half)
        float ks = 0.0f;
#pragma unroll
        for (int t = 0; t < 16; ++t) {
            int nt = __shfl(nb, t, 32);
            ks += P[(size_t)nt * PROJ + 64 + lane]; // gather k_loc rows (L2-resident)
        }
        float el = q * ks * INV2;
        float gl = Pn[96 + lane] * kg * INV2;
        float tot = es + el + gl;
#pragma unroll
        for (int off = 16; off > 0; off >>= 1) tot += __shfl_xor(tot, off, 32);
        float rcp = 1.0f / (0.001f + tot);
        float* Si = &S[wave][i * SSTR];
        Si[lane]      = es * rcp;
        Si[32 + lane] = el * rcp;
        Si[64 + lane] = gl * rcp;
    }
    __syncthreads();

    // Phase 2: out tile [16 x 512] = S[16 x 96] @ VcatT^T, 4 col-tiles at a time.
    const float* Sw = &S[wave][0];
    for (int gc = 0; gc < 8; ++gc) {
        v8f z = {0, 0, 0, 0, 0, 0, 0, 0};
        v8f acc[4];
#pragma unroll
        for (int g2 = 0; g2 < 4; ++g2) acc[g2] = z;
        for (int j = 0; j < 24; ++j) {
            v2f a = *(const v2f*)(Sw + m * SSTR + 4 * j + 2 * h);
#pragma unroll
            for (int g2 = 0; g2 < 4; ++g2) {
                int g = gc * 4 + g2;
                v2f b = *(const v2f*)(VT + (size_t)(16 * g + m) * KCAT + 4 * j + 2 * h);
                acc[g2] = __builtin_amdgcn_wmma_f32_16x16x4_f32(
                    false, a, false, b, (short)0, acc[g2], false, false);
            }
        }
#pragma unroll
        for (int g2 = 0; g2 < 4; ++g2) {
            int col = 16 * (gc * 4 + g2) + m;
            float bb = BN[col];
#pragma unroll
            for (int r = 0; r < 8; ++r) {
                int row = base + r + 8 * h;
                if (row < N_NODES) out[(size_t)row * D_IN + col] = acc[g2][r] + bb;
            }
        }
    }
}

// ---------------------------------------------------------------- launch
extern "C" void kernel_launch(void* const* d_in, const int* in_sizes, int n_in,
                              void* d_out, int out_size, void* d_ws, size_t ws_size,
                              hipStream_t stream) {
    (void)in_sizes; (void)n_in; (void)out_size; (void)ws_size;
    const int*   adj    = (const int*)  d_in[0];
    const float* x      = (const float*)d_in[1];
    const float* qk_ego = (const float*)d_in[2];
    const float* ve     = (const float*)d_in[3];
    const float* ql     = (const float*)d_in[4];
    const float* kl     = (const float*)d_in[5];
    const float* vl     = (const float*)d_in[6];
    const float* qg     = (const float*)d_in[7];
    const float* kgw    = (const float*)d_in[8];
    const float* vg     = (const float*)d_in[9];
    const float* bias   = (const float*)d_in[10];
    float* out = (float*)d_out;

    float* ws = (float*)d_ws;
    const size_t P_OFF  = 0;
    const size_t WC_OFF = P_OFF  + (size_t)N_NODES * PROJ;   // 12,800,000
    const size_t VT_OFF = WC_OFF + (size_t)PROJ * D_IN;      // + 65,536
    const size_t BN_OFF = VT_OFF + (size_t)D_IN * KCAT;      // + 49,152
    const size_t XB_OFF = BN_OFF + D_IN;
    const size_t KG_OFF = XB_OFF + D_IN;
    const size_t SM_OFF = KG_OFF + DB;
    float* P  = ws + P_OFF;
    float* WC = ws + WC_OFF;
    float* VT = ws + VT_OFF;
    float* BN = ws + BN_OFF;
    float* XB = ws + XB_OFF;
    float* KG = ws + KG_OFF;
    float* SM = ws + SM_OFF;

    Steamboat_sums_kernel<<<1, 256, 0, stream>>>(ql, qg, SM, XB);
    Steamboat_mean_kernel<<<dim3(2, 128), 256, 0, stream>>>(x, XB);
    {
        int total = PROJ * D_IN + D_IN * KCAT + D_IN;
        Steamboat_prep_kernel<<<(total + 255) / 256, 256, 0, stream>>>(
            qk_ego, ql, kl, qg, ve, vl, vg, bias, SM, WC, VT, BN);
    }
    Steamboat_kglob_kernel<<<1, 32, 0, stream>>>(XB, kgw, KG);

    const int nblk = (N_NODES + 127) / 128;   // 782 blocks, 8 waves * 16 nodes each
    Steamboat_proj_kernel<<<nblk, 256, 0, stream>>>(x, WC, P);
    Steamboat_final_kernel<<<nblk, 256, 0, stream>>>(adj, P, VT, BN, KG, out);
}